// SFCModel_41712722379521
// MI455X (gfx1250) — compile-verified
//
#include <hip/hip_runtime.h>

typedef __attribute__((ext_vector_type(2))) float v2f;
typedef __attribute__((ext_vector_type(8))) float v8f;

// One wave (32 lanes) computes 16 per-sample dot products via
// V_WMMA_F32_16X16X4_F32 accumulation over K=64 in 16 steps of K=4.
// A = 16 user rows (16x4 per step), B = 16 item rows as 4x16.
// diag(C) = per-sample dot products (full fp32 precision).
__global__ __launch_bounds__(256)
void sfc_wmma_kernel(const int* __restrict__ user,
                     const int* __restrict__ item,
                     const int* __restrict__ freq,
                     const int* __restrict__ idx_emb,
                     const float* __restrict__ bias_w,
                     const float* __restrict__ user_w,
                     const float* __restrict__ item_w,
                     const float* __restrict__ freq_w,
                     float* __restrict__ out,
                     int n)
{
    const int lane = threadIdx.x & 31;
    const int wave = threadIdx.x >> 5;
    const int s0   = (blockIdx.x * 8 + wave) * 16;   // 16 samples per wave

    // Sample this lane stages operand data for (lanes L and L+16 share sample L&15).
    const int  sl  = lane & 15;
    int        gs  = s0 + sl;
    const int  gsc = (gs < n) ? gs : (n - 1);        // clamp so EXEC stays all-ones

    const int uidx = user[gsc];
    const int iidx = item[gsc];
    const float* __restrict__ urow = user_w + (size_t)uidx * 64;
    const float* __restrict__ irow = item_w + (size_t)iidx * 64;

    // ISA layout for 32-bit 16x4 A (and 4x16 B):
    //   lanes 0-15  hold K = k0+0, k0+1  (VGPR0, VGPR1)
    //   lanes 16-31 hold K = k0+2, k0+3  (VGPR0, VGPR1)
    const int koff = (lane >> 4) * 2;

    v8f c = {};
#pragma unroll
    for (int k0 = 0; k0 < 64; k0 += 4) {
        v2f a = *(const v2f*)(urow + k0 + koff);
        v2f b = *(const v2f*)(irow + k0 + koff);
        // 8 args: (neg_a, A, neg_b, B, c_mod, C, reuse_a, reuse_b)
        c = __builtin_amdgcn_wmma_f32_16x16x4_f32(
                false, a, false, b, (short)0, c, false, false);
    }

    // Diagonal extraction per the 16x16 f32 C/D layout:
    //   diag(s), s<8 : VGPR s,   lane s
    //   diag(s), s>=8: VGPR s-8, lane s+16   (lanes 24..31)
    const bool active = (lane < 8) || (lane >= 24);
    const int  comp   = lane & 7;
    float d = c[0];
    d = (comp == 1) ? c[1] : d;
    d = (comp == 2) ? c[2] : d;
    d = (comp == 3) ? c[3] : d;
    d = (comp == 4) ? c[4] : d;
    d = (comp == 5) ? c[5] : d;
    d = (comp == 6) ? c[6] : d;
    d = (comp == 7) ? c[7] : d;

    if (active) {
        const int s = (lane < 8) ? lane : (lane - 16);   // 0..15
        const int g = s0 + s;
        if (g < n) {
            const float bias = bias_w[0];
            const int   fb   = freq[g];      // bin, 0..31
            const int   ft   = idx_emb[g];   // table, 0..49
            const float fe   = freq_w[ft * 32 + fb];  // freq_w: [50,32,1]
            out[g] = bias + d + fe;
        }
    }
}

extern "C" void kernel_launch(void* const* d_in, const int* in_sizes, int n_in,
                              void* d_out, int out_size, void* d_ws, size_t ws_size,
                              hipStream_t stream)
{
    // setup_inputs() order:
    // 0 user, 1 item, 2 freq, 3 idx_emb, 4 zero (unused),
    // 5 bias_w, 6 user_w, 7 item_w, 8 freq_w
    const int*   user    = (const int*)  d_in[0];
    const int*   item    = (const int*)  d_in[1];
    const int*   freq    = (const int*)  d_in[2];
    const int*   idx_emb = (const int*)  d_in[3];
    const float* bias_w  = (const float*)d_in[5];
    const float* user_w  = (const float*)d_in[6];
    const float* item_w  = (const float*)d_in[7];
    const float* freq_w  = (const float*)d_in[8];
    float*       out     = (float*)d_out;

    const int n = in_sizes[0];                 // BATCH
    const int samples_per_block = 8 * 16;      // 8 waves x 16 samples
    const int blocks = (n + samples_per_block - 1) / samples_per_block;

    sfc_wmma_kernel<<<blocks, 256, 0, stream>>>(
        user, item, freq, idx_emb, bias_w, user_w, item_w, freq_w, out, n);
}